// GATWithBatchNorm_8177617732166
// MI455X (gfx1250) — compile-verified
//
#include <hip/hip_runtime.h>
#include <math.h>

typedef __attribute__((ext_vector_type(2))) float v2f;
typedef __attribute__((ext_vector_type(8))) float v8f;

#define NEG_SLOPE 0.2f
#define EPS_BN 1e-5f
#define FLT_NEG_MAX -3.402823466e38f

// ---------------------------------------------------------------------------
// Utility kernels
// ---------------------------------------------------------------------------
__global__ void fill_f32(float* __restrict__ p, float v, int n) {
    int i = blockIdx.x * blockDim.x + threadIdx.x;
    if (i < n) p[i] = v;
}

__device__ __forceinline__ void atomicMaxFloat(float* addr, float val) {
    if (val >= 0.0f) atomicMax((int*)addr, __float_as_int(val));
    else             atomicMin((unsigned int*)addr, __float_as_uint(val));
}

// ---------------------------------------------------------------------------
// fp32 GEMM via V_WMMA_F32_16X16X4_F32.  D[n x NCOL] = A[n x K] * B[K x NCOL]
// One wave computes one 16x16 tile: block = (NCOL/16) waves, blockIdx = row tile.
// ---------------------------------------------------------------------------
template <int K, int NCOL>
__global__ void gemm_wmma(const float* __restrict__ A, const float* __restrict__ B,
                          float* __restrict__ D, int nrows) {
    const int lane = threadIdx.x & 31;
    const int wv   = threadIdx.x >> 5;        // column tile
    const int m    = lane & 15;
    const int rowBase = blockIdx.x * 16;
    int row = rowBase + m;
    if (row >= nrows) row = nrows - 1;        // clamp (stores are guarded)
    const int col   = wv * 16 + m;
    const int khalf = (lane >> 4) << 1;       // 0 for lanes 0-15, 2 for 16-31

    v8f c = {};
    #pragma unroll 4
    for (int k0 = 0; k0 < K; k0 += 4) {
        const int ka = k0 + khalf;
        v2f a, b;
        a.x = A[row * K + ka];
        a.y = A[row * K + ka + 1];
        b.x = B[ka * NCOL + col];
        b.y = B[(ka + 1) * NCOL + col];
        c = __builtin_amdgcn_wmma_f32_16x16x4_f32(false, a, false, b,
                                                  (short)0, c, false, false);
    }
    const int rOff = (lane >= 16) ? 8 : 0;
    if (rowBase + 16 <= nrows) {
        // Full tile (always taken for N % 16 == 0): unguarded stores, EXEC stays all-1.
        #pragma unroll
        for (int v = 0; v < 8; ++v)
            D[(rowBase + v + rOff) * NCOL + col] = c[v];
    } else {
        #pragma unroll
        for (int v = 0; v < 8; ++v) {
            int r = rowBase + v + rOff;
            if (r < nrows) D[r * NCOL + col] = c[v];
        }
    }
}

// ---------------------------------------------------------------------------
// Per-node attention logits: asrc[n,h] = sum_c h[n,h,c]*a_s[h,c]  (same for adst)
// ---------------------------------------------------------------------------
__global__ void attn_scores(const float* __restrict__ h, const float* __restrict__ a_s,
                            const float* __restrict__ a_d, float* __restrict__ asrc,
                            float* __restrict__ adst, int n, int heads, int ch) {
    int t = blockIdx.x * blockDim.x + threadIdx.x;
    if (t >= n * heads) return;
    int node = t / heads, hd = t % heads;
    const float* hp = h + (long)node * heads * ch + hd * ch;
    const float* as = a_s + hd * ch;
    const float* ad = a_d + hd * ch;
    float ss = 0.f, sd = 0.f;
    for (int c = 0; c < ch; c += 4) {
        float4 hv = *(const float4*)(hp + c);
        float4 av = *(const float4*)(as + c);
        float4 dv = *(const float4*)(ad + c);
        ss += hv.x*av.x + hv.y*av.y + hv.z*av.z + hv.w*av.w;
        sd += hv.x*dv.x + hv.y*dv.y + hv.z*dv.z + hv.w*dv.w;
    }
    asrc[t] = ss;
    adst[t] = sd;
}

__device__ __forceinline__ void edge_nodes(const int* __restrict__ ei, int E, int idx,
                                           int& s, int& d) {
    if (idx < E) { s = ei[idx]; d = ei[E + idx]; }
    else         { s = d = idx - E; }           // self loops appended
}

__device__ __forceinline__ float edge_logit(const float* asrc, const float* adst,
                                            int s, int d, int heads, int hd) {
    float e = asrc[s * heads + hd] + adst[d * heads + hd];
    return e > 0.f ? e : NEG_SLOPE * e;
}

// Pass 1: per-destination segment max
__global__ void edge_max(const int* __restrict__ ei, int E, int n, int heads,
                         const float* __restrict__ asrc, const float* __restrict__ adst,
                         float* __restrict__ mbuf) {
    int t = blockIdx.x * blockDim.x + threadIdx.x;
    int tot = (E + n) * heads;
    if (t >= tot) return;
    int hd = t % heads, idx = t / heads;
    int s, d; edge_nodes(ei, E, idx, s, d);
    float e = edge_logit(asrc, adst, s, d, heads, hd);
    atomicMaxFloat(&mbuf[d * heads + hd], e);
}

// Pass 2: segment sum of exp(e - m)
__global__ void edge_expsum(const int* __restrict__ ei, int E, int n, int heads,
                            const float* __restrict__ asrc, const float* __restrict__ adst,
                            const float* __restrict__ mbuf, float* __restrict__ sbuf) {
    int t = blockIdx.x * blockDim.x + threadIdx.x;
    int tot = (E + n) * heads;
    if (t >= tot) return;
    int hd = t % heads, idx = t / heads;
    int s, d; edge_nodes(ei, E, idx, s, d);
    float e = edge_logit(asrc, adst, s, d, heads, hd);
    atomicAdd(&sbuf[d * heads + hd], __expf(e - mbuf[d * heads + hd]));
}

// Pass 3: out[dst] += alpha * h[src]   (memory-bound core; prefetch the gather row)
__global__ void edge_scatter(const int* __restrict__ ei, int E, int n, int heads, int ch,
                             const float* __restrict__ asrc, const float* __restrict__ adst,
                             const float* __restrict__ mbuf, const float* __restrict__ sbuf,
                             const float* __restrict__ hfeat, float* __restrict__ outacc) {
    int t = blockIdx.x * blockDim.x + threadIdx.x;
    int tot = (E + n) * heads;
    if (t >= tot) return;
    int hd = t % heads, idx = t / heads;
    int s, d; edge_nodes(ei, E, idx, s, d);
    const float* hp = hfeat + (long)s * heads * ch + hd * ch;
    // Kick off the random gather early (global_prefetch); overlaps the scalar
    // logit/max/sum lookups below with the feature-row fetch.
    __builtin_prefetch(hp, 0, 3);
    __builtin_prefetch(hp + 32, 0, 3);
    float e = edge_logit(asrc, adst, s, d, heads, hd);
    float alpha = __expf(e - mbuf[d * heads + hd]) / (sbuf[d * heads + hd] + 1e-16f);
    float* op = outacc + (long)d * heads * ch + hd * ch;
    for (int c = 0; c < ch; c += 4) {
        float4 hv = *(const float4*)(hp + c);
        atomicAdd(op + c + 0, alpha * hv.x);
        atomicAdd(op + c + 1, alpha * hv.y);
        atomicAdd(op + c + 2, alpha * hv.z);
        atomicAdd(op + c + 3, alpha * hv.w);
    }
}

// ---------------------------------------------------------------------------
// BatchNorm: per-column sum / sumsq (block per column), then fused apply+ReLU
// ---------------------------------------------------------------------------
__global__ void bn_stats(const float* __restrict__ v, const float* __restrict__ bias,
                         float* __restrict__ sums, float* __restrict__ sqs,
                         int n, int ncol) {
    int col = blockIdx.x;
    float b = bias[col];
    float s = 0.f, q = 0.f;
    for (int r = threadIdx.x; r < n; r += blockDim.x) {
        float x = v[(long)r * ncol + col] + b;
        s += x; q += x * x;
    }
    __shared__ float sh[256], sh2[256];
    sh[threadIdx.x] = s; sh2[threadIdx.x] = q;
    __syncthreads();
    for (int off = blockDim.x >> 1; off > 0; off >>= 1) {
        if ((int)threadIdx.x < off) {
            sh[threadIdx.x]  += sh[threadIdx.x + off];
            sh2[threadIdx.x] += sh2[threadIdx.x + off];
        }
        __syncthreads();
    }
    if (threadIdx.x == 0) { sums[col] = sh[0]; sqs[col] = sh2[0]; }
}

__global__ void bn_apply_relu(const float* __restrict__ v, const float* __restrict__ bias,
                              const float* __restrict__ sums, const float* __restrict__ sqs,
                              const float* __restrict__ gamma, const float* __restrict__ beta,
                              float* __restrict__ out, int n, int ncol) {
    int t = blockIdx.x * blockDim.x + threadIdx.x;
    if (t >= n * ncol) return;
    int col = t % ncol;
    float inv_n = 1.0f / (float)n;
    float mu  = sums[col] * inv_n;
    float var = sqs[col] * inv_n - mu * mu;
    float x = v[t] + bias[col];
    float y = gamma[col] * (x - mu) * rsqrtf(var + EPS_BN) + beta[col];
    out[t] = y > 0.f ? y : 0.f;
}

// ---------------------------------------------------------------------------
// Global mean pool + final FC
// ---------------------------------------------------------------------------
__global__ void pool_sum(const float* __restrict__ h, const int* __restrict__ batch,
                         float* __restrict__ pooled, float* __restrict__ counts,
                         int n, int ch) {
    int t = blockIdx.x * blockDim.x + threadIdx.x;
    if (t >= n * ch) return;
    int node = t / ch, c = t % ch;
    int g = batch[node];
    atomicAdd(&pooled[g * ch + c], h[t]);
    if (c == 0) atomicAdd(&counts[g], 1.0f);
}

__global__ void final_fc(const float* __restrict__ pooled, const float* __restrict__ counts,
                         const float* __restrict__ Wfc, const float* __restrict__ bfc,
                         float* __restrict__ out, int G, int ch, int od) {
    int t = blockIdx.x * blockDim.x + threadIdx.x;
    if (t >= G * od) return;
    int g = t / od, o = t % od;
    float cnt = counts[g];
    cnt = cnt > 1.f ? cnt : 1.f;
    float inv = 1.f / cnt;
    float acc = bfc[o];
    for (int c = 0; c < ch; ++c)
        acc += pooled[g * ch + c] * inv * Wfc[c * od + o];
    out[t] = acc;
}

// ---------------------------------------------------------------------------
// Launch
// ---------------------------------------------------------------------------
static inline int cdiv(long a, long b) { return (int)((a + b - 1) / b); }

extern "C" void kernel_launch(void* const* d_in, const int* in_sizes, int n_in,
                              void* d_out, int out_size, void* d_ws, size_t ws_size,
                              hipStream_t stream) {
    const float* x    = (const float*)d_in[0];
    const int*   ei   = (const int*)  d_in[1];   // [2,E] row-major: src then dst
    const int*   batch= (const int*)  d_in[2];
    const float* W1   = (const float*)d_in[3];
    const float* a_s1 = (const float*)d_in[4];
    const float* a_d1 = (const float*)d_in[5];
    const float* b1   = (const float*)d_in[6];
    const float* g1   = (const float*)d_in[7];
    const float* be1  = (const float*)d_in[8];
    const float* W2   = (const float*)d_in[9];
    const float* a_s2 = (const float*)d_in[10];
    const float* a_d2 = (const float*)d_in[11];
    const float* b2   = (const float*)d_in[12];
    const float* g2   = (const float*)d_in[13];
    const float* be2  = (const float*)d_in[14];
    const float* Wfc  = (const float*)d_in[15];
    const float* bfc  = (const float*)d_in[16];
    float* out = (float*)d_out;

    const int n   = in_sizes[2];          // 50000 nodes
    const int E   = in_sizes[1] / 2;      // 500000 edges
    const int H1  = 4, CH = 64, HC = 256;
    const int OD  = in_sizes[16];         // 64
    const int G   = out_size / OD;        // 512
    const int Etot = E + n;               // with self loops

    // ---- workspace bump allocator -------------------------------------------
    char* p = (char*)d_ws;
    auto alloc = [&](size_t bytes) -> float* {
        float* r = (float*)p;
        p += (bytes + 255) & ~(size_t)255;
        return r;
    };
    float* h1    = alloc((size_t)n * HC * 4);   // layer1 features / layer2 input
    float* out1  = alloc((size_t)n * HC * 4);   // layer1 aggregation
    float* asrc1 = alloc((size_t)n * H1 * 4);
    float* adst1 = alloc((size_t)n * H1 * 4);
    float* m1    = alloc((size_t)n * H1 * 4);
    float* s1    = alloc((size_t)n * H1 * 4);
    float* bns1  = alloc(HC * 4);
    float* bnq1  = alloc(HC * 4);
    float* h2    = alloc((size_t)n * CH * 4);
    float* out2  = alloc((size_t)n * CH * 4);
    float* asrc2 = alloc((size_t)n * 4);
    float* adst2 = alloc((size_t)n * 4);
    float* m2    = alloc((size_t)n * 4);
    float* s2    = alloc((size_t)n * 4);
    float* bns2  = alloc(CH * 4);
    float* bnq2  = alloc(CH * 4);
    float* pooled= alloc((size_t)G * CH * 4);
    float* counts= alloc((size_t)G * 4);

    const int TB = 256;

    // ---- re-init accumulators every call (graph-replay safe) ----------------
    fill_f32<<<cdiv((long)n * HC, TB), TB, 0, stream>>>(out1, 0.f, n * HC);
    fill_f32<<<cdiv((long)n * H1, TB), TB, 0, stream>>>(m1, FLT_NEG_MAX, n * H1);
    fill_f32<<<cdiv((long)n * H1, TB), TB, 0, stream>>>(s1, 0.f, n * H1);
    fill_f32<<<cdiv((long)n * CH, TB), TB, 0, stream>>>(out2, 0.f, n * CH);
    fill_f32<<<cdiv(n, TB), TB, 0, stream>>>(m2, FLT_NEG_MAX, n);
    fill_f32<<<cdiv(n, TB), TB, 0, stream>>>(s2, 0.f, n);
    fill_f32<<<cdiv((long)G * CH, TB), TB, 0, stream>>>(pooled, 0.f, G * CH);
    fill_f32<<<cdiv(G, TB), TB, 0, stream>>>(counts, 0.f, G);

    // ---- Layer 1 ------------------------------------------------------------
    gemm_wmma<128, 256><<<cdiv(n, 16), 512, 0, stream>>>(x, W1, h1, n);
    attn_scores<<<cdiv((long)n * H1, TB), TB, 0, stream>>>(h1, a_s1, a_d1, asrc1, adst1, n, H1, CH);

    int egrid = cdiv((long)Etot * H1, TB);
    edge_max    <<<egrid, TB, 0, stream>>>(ei, E, n, H1, asrc1, adst1, m1);
    edge_expsum <<<egrid, TB, 0, stream>>>(ei, E, n, H1, asrc1, adst1, m1, s1);
    edge_scatter<<<egrid, TB, 0, stream>>>(ei, E, n, H1, CH, asrc1, adst1, m1, s1, h1, out1);

    bn_stats<<<HC, 256, 0, stream>>>(out1, b1, bns1, bnq1, n, HC);
    bn_apply_relu<<<cdiv((long)n * HC, TB), TB, 0, stream>>>(out1, b1, bns1, bnq1, g1, be1, h1, n, HC);

    // ---- Layer 2 ------------------------------------------------------------
    gemm_wmma<256, 64><<<cdiv(n, 16), 128, 0, stream>>>(h1, W2, h2, n);
    attn_scores<<<cdiv(n, TB), TB, 0, stream>>>(h2, a_s2, a_d2, asrc2, adst2, n, 1, CH);

    int egrid2 = cdiv(Etot, TB);
    edge_max    <<<egrid2, TB, 0, stream>>>(ei, E, n, 1, asrc2, adst2, m2);
    edge_expsum <<<egrid2, TB, 0, stream>>>(ei, E, n, 1, asrc2, adst2, m2, s2);
    edge_scatter<<<egrid2, TB, 0, stream>>>(ei, E, n, 1, CH, asrc2, adst2, m2, s2, h2, out2);

    bn_stats<<<CH, 256, 0, stream>>>(out2, b2, bns2, bnq2, n, CH);
    bn_apply_relu<<<cdiv((long)n * CH, TB), TB, 0, stream>>>(out2, b2, bns2, bnq2, g2, be2, h2, n, CH);

    // ---- Pool + FC ----------------------------------------------------------
    pool_sum<<<cdiv((long)n * CH, TB), TB, 0, stream>>>(h2, batch, pooled, counts, n, CH);
    final_fc<<<cdiv((long)G * OD, TB), TB, 0, stream>>>(pooled, counts, Wfc, bfc, out, G, CH, OD);
}